// make_dense_gcn_layer_88691074662579
// MI455X (gfx1250) — compile-verified
//
#include <hip/hip_runtime.h>
#include <hip/hip_bf16.h>

typedef __attribute__((ext_vector_type(2))) float v2f;
typedef __attribute__((ext_vector_type(8))) float v8f;

#define TM 64
#define TN 64
#define TK 32
#define NNODES 1024
#define NEGC (-9.0e15f)

// ---- gfx1250 async global->LDS copy path (ASYNCcnt-tracked) ----------------
#if defined(__has_builtin)
#if __has_builtin(__builtin_amdgcn_global_load_async_to_lds_b128) && \
    __has_builtin(__builtin_amdgcn_s_wait_asynccnt)
#define USE_ASYNC_LDS 1
#endif
#endif

typedef int v4i_vs __attribute__((vector_size(16)));
typedef __attribute__((address_space(1))) v4i_vs glob_v4i;
typedef __attribute__((address_space(3))) v4i_vs lds_v4i;

__device__ __forceinline__ void cp16_to_lds(const float* g, float* l)
{
#ifdef USE_ASYNC_LDS
  // per-lane 16B async copy: emits global_load_async_to_lds_b128
  __builtin_amdgcn_global_load_async_to_lds_b128(
      (glob_v4i*)(uintptr_t)g, (lds_v4i*)(uintptr_t)l, 0, 0);
#else
  *(float4*)l = *(const float4*)g;
#endif
}

__device__ __forceinline__ void tiles_ready()
{
#ifdef USE_ASYNC_LDS
  __builtin_amdgcn_s_wait_asynccnt(0);
#endif
  __syncthreads();
}

__device__ __forceinline__ float leakyf(float x) { return x > 0.0f ? x : 0.01f * x; }

// ---------------------------------------------------------------------------
// Generic fp32 WMMA GEMM: C[M,N] = act(A[M,K] @ B[K,N] + bias) (+resid)
// 64x64 tile, 256 threads = 8 waves, each wave owns a 16x32 strip (2 frags).
// M,N multiples of 64; K multiple of 32. Output written at column offset ccol.
// ---------------------------------------------------------------------------
__global__ __launch_bounds__(256)
void gemm_wmma_f32(const float* __restrict__ A, int lda,
                   const float* __restrict__ B, int ldb,
                   float* __restrict__ C, int ldc, int ccol,
                   int M, int N, int Kdim,
                   const float* __restrict__ bias,
                   const float* __restrict__ resid, int ldr,
                   int do_leaky)
{
  __shared__ __align__(16) float As[TM][TK];
  __shared__ __align__(16) float Bs[TK][TN];

  const int tid  = threadIdx.x;
  const int wave = tid >> 5;
  const int lane = tid & 31;
  const int lrow = lane & 15;
  const int lhi  = lane >> 4;
  const int m0 = blockIdx.x * TM;
  const int n0 = blockIdx.y * TN;
  const int wm = (wave >> 1) * 16;   // wave M offset inside tile (0..48)
  const int wn = (wave & 1) * 32;    // wave N offset inside tile (0 or 32)

  v8f c0 = {};
  v8f c1 = {};

  for (int k0 = 0; k0 < Kdim; k0 += TK) {
    // Cooperative tile loads: 64x32 A and 32x64 B, 2x16B per thread each,
    // issued as async global->LDS copies (no VGPR staging).
    #pragma unroll
    for (int r = 0; r < 2; ++r) {
      int idx = tid + r * 256;
      int ar = idx >> 3, ac = (idx & 7) << 2;
      cp16_to_lds(&A[(size_t)(m0 + ar) * lda + k0 + ac], &As[ar][ac]);
      int br = idx >> 4, bc = (idx & 15) << 2;
      cp16_to_lds(&B[(size_t)(k0 + br) * ldb + n0 + bc], &Bs[br][bc]);
    }
    if (k0 + TK < Kdim) {   // hint next A tile toward L2 (global_prefetch_b8)
      __builtin_prefetch(&A[(size_t)(m0 + (tid >> 2)) * lda + k0 + TK], 0, 1);
    }
    tiles_ready();

    #pragma unroll
    for (int kk = 0; kk < TK; kk += 4) {
      const int kb = kk + lhi * 2;
      // A 16x4 frag: lanes carry M, v0/v1 carry {K+0,K+1 | K+2,K+3}
      v2f a = *(const v2f*)&As[wm + lrow][kb];
      // B 4x16 frags: lanes carry N, v0/v1 carry {K+0,K+1 | K+2,K+3}
      v2f b0, b1;
      b0.x = Bs[kb][wn + lrow];       b0.y = Bs[kb + 1][wn + lrow];
      b1.x = Bs[kb][wn + 16 + lrow];  b1.y = Bs[kb + 1][wn + 16 + lrow];
      c0 = __builtin_amdgcn_wmma_f32_16x16x4_f32(false, a, false, b0,
                                                 (short)0, c0, false, false);
      c1 = __builtin_amdgcn_wmma_f32_16x16x4_f32(false, a, false, b1,
                                                 (short)0, c1, false, false);
    }
    __syncthreads();
  }

  // C 16x16 frag layout: vgpr r -> M = r (lanes 0-15) / r+8 (lanes 16-31)
  #pragma unroll
  for (int r = 0; r < 8; ++r) {
    int row = m0 + wm + r + lhi * 8;
    int gc0 = n0 + wn + lrow;
    int gc1 = gc0 + 16;
    float v0 = c0[r], v1 = c1[r];
    if (bias) { v0 += bias[gc0]; v1 += bias[gc1]; }
    if (do_leaky) { v0 = leakyf(v0); v1 = leakyf(v1); }
    if (resid) {
      v0 += resid[(size_t)row * ldr + gc0];
      v1 += resid[(size_t)row * ldr + gc1];
    }
    C[(size_t)row * ldc + ccol + gc0] = v0;
    C[(size_t)row * ldc + ccol + gc1] = v1;
  }
}

// ---------------------------------------------------------------------------
// Attention-apply GEMM: C = att @ h, att tile materialized in LDS from
// adj + collapsed logits (rowEa/rowEb for i<512, colG for i>=512) and
// per-row softmax stats (rowM, rowInvZ). M=K=1024, N=256.
// ---------------------------------------------------------------------------
__global__ __launch_bounds__(256)
void att_gemm_wmma(const int* __restrict__ adj, int head,
                   const float* __restrict__ rowEa,
                   const float* __restrict__ rowEb,
                   const float* __restrict__ colG,
                   const float* __restrict__ rowM,
                   const float* __restrict__ rowInvZ,
                   const float* __restrict__ B, int ldb,
                   float* __restrict__ C, int ldc, int ccol)
{
  __shared__ __align__(16) float As[TM][TK];
  __shared__ __align__(16) float Bs[TK][TN];

  const int tid  = threadIdx.x;
  const int wave = tid >> 5;
  const int lane = tid & 31;
  const int lrow = lane & 15;
  const int lhi  = lane >> 4;
  const int m0 = blockIdx.x * TM;
  const int n0 = blockIdx.y * TN;
  const int wm = (wave >> 1) * 16;
  const int wn = (wave & 1) * 32;

  v8f c0 = {};
  v8f c1 = {};

  for (int k0 = 0; k0 < NNODES; k0 += TK) {
    // B tile via async global->LDS copy.
    #pragma unroll
    for (int r = 0; r < 2; ++r) {
      int idx = tid + r * 256;
      int br = idx >> 4, bc = (idx & 15) << 2;
      cp16_to_lds(&B[(size_t)(k0 + br) * ldb + n0 + bc], &Bs[br][bc]);
    }
    // Build 64x32 attention tile in LDS: 8 entries per thread, coalesced in j.
    #pragma unroll
    for (int r = 0; r < 8; ++r) {
      int idx = tid + r * 256;
      int ar = idx >> 5, ac = idx & 31;
      int i = m0 + ar, j = k0 + ac;
      bool mk = adj[((size_t)i * 2 + head) * NNODES + j] > 0;
      float e = (i < 512) ? (j < 512 ? rowEa[i] : rowEb[i]) : colG[j];
      float val = mk ? e : NEGC;
      As[ar][ac] = __expf(val - rowM[i]) * rowInvZ[i];
    }
    tiles_ready();

    #pragma unroll
    for (int kk = 0; kk < TK; kk += 4) {
      const int kb = kk + lhi * 2;
      v2f a = *(const v2f*)&As[wm + lrow][kb];
      v2f b0, b1;
      b0.x = Bs[kb][wn + lrow];       b0.y = Bs[kb + 1][wn + lrow];
      b1.x = Bs[kb][wn + 16 + lrow];  b1.y = Bs[kb + 1][wn + 16 + lrow];
      c0 = __builtin_amdgcn_wmma_f32_16x16x4_f32(false, a, false, b0,
                                                 (short)0, c0, false, false);
      c1 = __builtin_amdgcn_wmma_f32_16x16x4_f32(false, a, false, b1,
                                                 (short)0, c1, false, false);
    }
    __syncthreads();
  }

  #pragma unroll
  for (int r = 0; r < 8; ++r) {
    int row = m0 + wm + r + lhi * 8;
    int gc0 = n0 + wn + lrow;
    int gc1 = gc0 + 16;
    C[(size_t)row * ldc + ccol + gc0] = c0[r];
    C[(size_t)row * ldc + ccol + gc1] = c1[r];
  }
}

// ---------------------------------------------------------------------------
// p1[t] = emb[t,:] . a[0:32];  p2[t] = emb[t,:] . a[32:64]   (t = 0..255)
// ---------------------------------------------------------------------------
__global__ __launch_bounds__(256)
void emb_proj(const float* __restrict__ emb, const float* __restrict__ avec,
              float* __restrict__ p)
{
  int t = threadIdx.x;
  const float* er = emb + (size_t)t * 32;
  float a = 0.0f, b = 0.0f;
  #pragma unroll
  for (int c = 0; c < 32; ++c) {
    float e = er[c];
    a += e * avec[c];
    b += e * avec[32 + c];
  }
  p[t] = a;
  p[256 + t] = b;
}

// ---------------------------------------------------------------------------
// s1[row] = h[row,:] . p1 ; s2[row] = h[row,:] . p2  — one wave32 per row
// ---------------------------------------------------------------------------
__global__ __launch_bounds__(32)
void node_scores(const float* __restrict__ h, const float* __restrict__ p,
                 float* __restrict__ s1, float* __restrict__ s2)
{
  int row = blockIdx.x;
  int lane = threadIdx.x;
  const float* hr = h + (size_t)row * 256;
  float a = 0.0f, b = 0.0f;
  #pragma unroll
  for (int c = lane; c < 256; c += 32) {
    float hv = hr[c];
    a += hv * p[c];
    b += hv * p[256 + c];
  }
  #pragma unroll
  for (int off = 16; off > 0; off >>= 1) {
    a += __shfl_down(a, off, 32);
    b += __shfl_down(b, off, 32);
  }
  if (lane == 0) { s1[row] = a; s2[row] = b; }
}

// ---------------------------------------------------------------------------
// Collapsed logits from the interleaved concat/reshape:
//   rowEa[i] = leaky(s1[2i]+s2[2i])       (rows i<512, cols j<512; also colG[j<512])
//   rowEb[i] = leaky(s1[2i+1]+s2[2i+1])   (rows i<512, cols j>=512)
//   colG[512+i] = leaky(s1[2i]+s2[2i+1])  (rows i>=512, cols j>=512)
// ---------------------------------------------------------------------------
__global__ __launch_bounds__(256)
void build_logits(const float* __restrict__ s1, const float* __restrict__ s2,
                  float* __restrict__ ea, float* __restrict__ eb,
                  float* __restrict__ cg)
{
  int i = blockIdx.x * 256 + threadIdx.x;
  if (i < 512) {
    float la = leakyf(s1[2 * i] + s2[2 * i]);
    float lb = leakyf(s1[2 * i + 1] + s2[2 * i + 1]);
    ea[i] = la;
    eb[i] = lb;
    cg[i] = la;
    cg[512 + i] = leakyf(s1[2 * i] + s2[2 * i + 1]);
  }
}

// ---------------------------------------------------------------------------
// Per-row masked softmax stats: rowM = max_j val(i,j), rowInvZ = 1/sum exp.
// One 256-thread block per row.
// ---------------------------------------------------------------------------
__global__ __launch_bounds__(256)
void att_stats(const int* __restrict__ adj, int head,
               const float* __restrict__ rowEa, const float* __restrict__ rowEb,
               const float* __restrict__ colG,
               float* __restrict__ rowM, float* __restrict__ rowInvZ)
{
  __shared__ float red[256];
  int i = blockIdx.x;
  int tid = threadIdx.x;
  float ef = 0.0f, es = 0.0f;
  if (i < 512) { ef = rowEa[i]; es = rowEb[i]; }

  const int* arow = adj + ((size_t)i * 2 + head) * NNODES;

  float m = -3.4e38f;
  for (int j = tid; j < NNODES; j += 256) {
    bool mk = arow[j] > 0;
    float e = (i < 512) ? (j < 512 ? ef : es) : colG[j];
    m = fmaxf(m, mk ? e : NEGC);
  }
  red[tid] = m;
  __syncthreads();
  for (int s = 128; s > 0; s >>= 1) {
    if (tid < s) red[tid] = fmaxf(red[tid], red[tid + s]);
    __syncthreads();
  }
  m = red[0];
  __syncthreads();

  float z = 0.0f;
  for (int j = tid; j < NNODES; j += 256) {
    bool mk = arow[j] > 0;
    float e = (i < 512) ? (j < 512 ? ef : es) : colG[j];
    z += __expf((mk ? e : NEGC) - m);
  }
  red[tid] = z;
  __syncthreads();
  for (int s = 128; s > 0; s >>= 1) {
    if (tid < s) red[tid] += red[tid + s];
    __syncthreads();
  }
  if (tid == 0) {
    rowM[i] = m;
    rowInvZ[i] = 1.0f / red[0];
  }
}

// ---------------------------------------------------------------------------
// Strided column copy (float4): dst[row, coloff+c] = src[row, c], 1024 rows.
// ---------------------------------------------------------------------------
__global__ __launch_bounds__(256)
void copy_cols(const float* __restrict__ src, int lds_,
               float* __restrict__ dst, int ldd, int cols4, int coloff)
{
  int idx = blockIdx.x * 256 + threadIdx.x;
  int r = idx / cols4;
  int c = (idx % cols4) << 2;
  if (r < NNODES) {
    *(float4*)&dst[(size_t)r * ldd + coloff + c] =
        *(const float4*)&src[(size_t)r * lds_ + c];
  }
}

// ---------------------------------------------------------------------------
extern "C" void kernel_launch(void* const* d_in, const int* in_sizes, int n_in,
                              void* d_out, int out_size, void* d_ws, size_t ws_size,
                              hipStream_t stream)
{
  const float* V    = (const float*)d_in[0];
  const int*   adj  = (const int*)d_in[1];
  const float* Ws[3] = {(const float*)d_in[2], (const float*)d_in[3],
                        (const float*)d_in[4]};
  const float* embs  = (const float*)d_in[5];
  const float* avecs = (const float*)d_in[6];
  const float* sqw   = (const float*)d_in[7];
  const float* sqb   = (const float*)d_in[8];
  float* out = (float*)d_out;
  float* ws  = (float*)d_ws;

  // Workspace layout (floats): ~9.5 MB total
  float* X  = ws;                       // 1024 x 768 growing feature concat
  float* H  = X + 1024 * 768;           // 3 x 1024 x 256 head outputs h_k
  float* O  = H + 3 * 1024 * 256;       // 1024 x 768 per-layer head concat
  float* P  = O + 1024 * 768;           // 2 heads x (p1[256], p2[256])
  float* S  = P + 2 * 512;              // 2 heads x (s1[1024], s2[1024])
  float* E  = S + 2 * 2048;             // 2 heads x (ea[512], eb[512], cg[1024])
  float* MZ = E + 2 * 2048;             // 2 heads x (rowM[1024], rowInvZ[1024])

  // X[:, 0:256] = V
  copy_cols<<<256, 256, 0, stream>>>(V, 256, X, 768, 64, 0);

  for (int l = 0; l < 3; ++l) {
    const int inf = 256 * (l + 1);

    // h_k = X @ W[l][k]  (k = 0,1,2)
    for (int k = 0; k < 3; ++k) {
      gemm_wmma_f32<<<dim3(16, 4), 256, 0, stream>>>(
          X, 768, Ws[l] + (size_t)k * inf * 256, 256,
          H + (size_t)k * 262144, 256, 0,
          1024, 256, inf, nullptr, nullptr, 0, 0);
    }

    // masked attention heads k = 0,1
    for (int k = 0; k < 2; ++k) {
      const float* emb = embs + (size_t)(l * 3 + k) * 256 * 32;
      const float* av  = avecs + (size_t)(l * 3 + k) * 64;
      float* p  = P + k * 512;
      float* s1 = S + k * 2048;
      float* s2 = s1 + 1024;
      float* ea = E + k * 2048;
      float* eb = ea + 512;
      float* cg = ea + 1024;
      float* rm = MZ + k * 2048;
      float* rz = rm + 1024;

      emb_proj<<<1, 256, 0, stream>>>(emb, av, p);
      node_scores<<<1024, 32, 0, stream>>>(H + (size_t)k * 262144, p, s1, s2);
      build_logits<<<2, 256, 0, stream>>>(s1, s2, ea, eb, cg);
      att_stats<<<1024, 256, 0, stream>>>(adj, k, ea, eb, cg, rm, rz);
      att_gemm_wmma<<<dim3(16, 4), 256, 0, stream>>>(
          adj, k, ea, eb, cg, rm, rz,
          H + (size_t)k * 262144, 256,
          O, 768, k * 256);
    }

    // identity head: att = I  =>  out_2 = h_2
    copy_cols<<<256, 256, 0, stream>>>(H + 2 * 262144, 256, O, 768, 64, 512);

    // squeeze: leaky(O @ sq_w[l] + sq_b[l]) (+ V residual on layer 0)
    float* Cdst = (l < 2) ? X : out;
    int ldcd    = (l < 2) ? 768 : 256;
    int coff    = (l < 2) ? 256 * (l + 1) : 0;
    const float* resid = (l == 0) ? X : nullptr;  // X cols 0:256 == V
    gemm_wmma_f32<<<dim3(16, 4), 256, 0, stream>>>(
        O, 768, sqw + (size_t)l * 768 * 256, 256,
        Cdst, ldcd, coff,
        1024, 256, 768, sqb + l * 256, resid, 768, 1);
  }
}